// TemporalGAT_19146964205623
// MI455X (gfx1250) — compile-verified
//
#include <hip/hip_runtime.h>
#include <math.h>

typedef __attribute__((ext_vector_type(2))) float v2f;
typedef __attribute__((ext_vector_type(8))) float v8f;

#define NB    32
#define CCH   64
#define TT    300
#define VV    25
#define NV    (NB*VV)       // 800 sequences
#define MROWS (NV*TT)       // 240000 rows of the big GEMM
#define TP    150           // selected rows (stride 2)
#define KHALF 4             // kernel//2

// ---------------------------------------------------------------------------
// Kernel 1: xw[m][c] = sum_k xv[m][k] * W[k][c]
//   xv[m][k] = x[n][k][t][v] with m = (n*25+v)*300 + t
//   V_WMMA_F32_16X16X4_F32 (fp32 matrix core, matches reference precision).
//   One wave per 16-row M tile computing all four 16-wide N tiles (A loads
//   amortized 4x). W staged in LDS as K-adjacent float2 pairs so every B
//   fragment is one aligned ds_load_b64 straight into an even VGPR pair.
// ---------------------------------------------------------------------------
__global__ void __launch_bounds__(128)
tgat_xw_wmma(const float* __restrict__ x, const float* __restrict__ W,
             float* __restrict__ xw) {
  // sWp[kp*64 + c] = { W[2kp][c], W[2kp+1][c] }
  __shared__ v2f sWp[32 * 64];
  for (int i = threadIdx.x; i < 32 * 64; i += 128) {
    const int kp  = i >> 6;
    const int col = i & 63;
    v2f p;
    p.x = W[(2 * kp) * 64 + col];
    p.y = W[(2 * kp + 1) * 64 + col];
    sWp[i] = p;
  }
  __syncthreads();

  const int lane  = threadIdx.x & 31;
  const int wave  = threadIdx.x >> 5;   // 4 waves -> 4 M tiles per block
  const int half  = lane >> 4;          // 0/1: K sub-pair per ISA A layout
  const int mloc  = lane & 15;
  const int mbase = blockIdx.x * 64 + wave * 16;

  // A-fragment row for this lane
  const int m  = mbase + mloc;
  const int nv = m / TT;
  const int t  = m - nv * TT;
  const int n  = nv / VV;
  const int v  = nv - n * VV;
  // x index(n,c,t,v) = ((n*64 + c)*300 + t)*25 + v ; c stride = 7500
  const float* xb = x + (size_t)n * (CCH * TT * VV) + (size_t)t * VV + v;

  v8f acc0 = {}, acc1 = {}, acc2 = {}, acc3 = {};
  for (int k = 0; k < 64; k += 4) {
    const int k0 = k + 2 * half;        // always even
    // A 16x4 layout: VGPR0 = K=k0 (lanes 0-15) / K=k0+2 (lanes 16-31); VGPR1 = +1
    v2f a;
    a.x = xb[(size_t)k0 * (TT * VV)];
    a.y = xb[(size_t)(k0 + 1) * (TT * VV)];
    // B 4x16 layout: {K=k0,K=k0+1} pair per lane, one b64 LDS load each
    const v2f* wp = sWp + (k0 >> 1) * 64 + mloc;
    const v2f b0 = wp[0];
    const v2f b1 = wp[16];
    const v2f b2 = wp[32];
    const v2f b3 = wp[48];
    acc0 = __builtin_amdgcn_wmma_f32_16x16x4_f32(false, a, false, b0, (short)0, acc0, false, false);
    acc1 = __builtin_amdgcn_wmma_f32_16x16x4_f32(false, a, false, b1, (short)0, acc1, false, false);
    acc2 = __builtin_amdgcn_wmma_f32_16x16x4_f32(false, a, false, b2, (short)0, acc2, false, false);
    acc3 = __builtin_amdgcn_wmma_f32_16x16x4_f32(false, a, false, b3, (short)0, acc3, false, false);
  }

  // D 16x16 layout: VGPR r holds M = r + 8*half, N = lane&15 (+16 per tile)
#pragma unroll
  for (int r = 0; r < 8; ++r) {
    float* orow = xw + (size_t)(mbase + r + 8 * half) * 64 + mloc;
    orow[0]  = acc0[r];
    orow[16] = acc1[r];
    orow[32] = acc2[r];
    orow[48] = acc3[r];
  }
}

// ---------------------------------------------------------------------------
// Kernel 2: banded attention per sequence.
//   Mask multiplies scores by 0 outside |t-s|<=4, so softmax still sees
//   exp(0)=1 there:
//   out = (rowsum + sum_band (e^s - 1)*xw[s]) / (300 + sum_band (e^s - 1))
// ---------------------------------------------------------------------------
__global__ void __launch_bounds__(256)
tgat_attn(const float* __restrict__ xw, const float* __restrict__ alpha,
          const float* __restrict__ phi, float* __restrict__ out) {
  __shared__ float sx1[TP * 16];   // x1 at selected rows only
  __shared__ float sx2[TT * 16];   // x2 at all rows
  __shared__ float srow[64];       // column sums of xw over all 300 rows
  __shared__ float se[TP * 9];     // exp(score)-1 inside band (0 if s OOB)
  __shared__ float sD[TP];         // softmax denominators

  const int nv = blockIdx.x;
  const int n  = nv / VV;
  const int v  = nv - n * VV;
  const float* xwB = xw + (size_t)nv * TT * 64;
  const int tid = threadIdx.x;

  // x2 for every row; x1 only for even rows (the stride-2 selection)
  for (int idx = tid; idx < TT * 16; idx += 256) {
    const int s = idx >> 4;
    const int h = idx & 15;
    const float* xr = xwB + (size_t)s * 64;
    const float* pr = phi + h * 64;
    float a2 = 0.f;
    for (int c = 0; c < 64; ++c) a2 += pr[c] * xr[c];
    sx2[idx] = a2;
    if ((s & 1) == 0) {
      const float* ar = alpha + h * 64;
      float a1 = 0.f;
      for (int c = 0; c < 64; ++c) a1 += ar[c] * xr[c];
      sx1[(s >> 1) * 16 + h] = a1;
    }
  }
  // column sums (64 columns)
  for (int c = tid; c < 64; c += 256) {
    float accv = 0.f;
    for (int s = 0; s < TT; ++s) accv += xwB[(size_t)s * 64 + c];
    srow[c] = accv;
  }
  __syncthreads();

  // band scores -> exp(score) - 1
  for (int idx = tid; idx < TP * 9; idx += 256) {
    const int tp = idx / 9;
    const int j  = idx - tp * 9;
    const int s  = 2 * tp - KHALF + j;
    float e = 0.f;
    if ((unsigned)s < (unsigned)TT) {
      float sc = 0.f;
      for (int h = 0; h < 16; ++h) sc += sx1[tp * 16 + h] * sx2[s * 16 + h];
      e = __expf(sc) - 1.f;
    }
    se[idx] = e;
  }
  __syncthreads();

  // denominators: 300 out-of-band "exp(0)" terms folded in analytically
  for (int tp = tid; tp < TP; tp += 256) {
    float D = (float)TT;
    for (int j = 0; j < 9; ++j) D += se[tp * 9 + j];
    sD[tp] = D;
  }
  __syncthreads();

  // outputs, stored in (N, C, Tp, V) order
  for (int idx = tid; idx < TP * 64; idx += 256) {
    const int tp = idx >> 6;
    const int c  = idx & 63;
    float accv = srow[c];
#pragma unroll
    for (int j = 0; j < 9; ++j) {
      int s = 2 * tp - KHALF + j;
      s = s < 0 ? 0 : (s > TT - 1 ? TT - 1 : s);   // se==0 when clamped
      accv += se[tp * 9 + j] * xwB[(size_t)s * 64 + c];
    }
    out[(((size_t)n * 64 + c) * TP + tp) * VV + v] = accv / sD[tp];
  }
}

extern "C" void kernel_launch(void* const* d_in, const int* in_sizes, int n_in,
                              void* d_out, int out_size, void* d_ws, size_t ws_size,
                              hipStream_t stream) {
  const float* x     = (const float*)d_in[0];
  const float* W     = (const float*)d_in[1];
  const float* alpha = (const float*)d_in[2];
  const float* phi   = (const float*)d_in[3];
  // d_in[4] = kernel(9), d_in[5] = stride(2): baked in as compile-time consts.

  float* xw = (float*)d_ws;  // 240000 * 64 * 4 B = 61.44 MB scratch

  tgat_xw_wmma<<<MROWS / 64, 128, 0, stream>>>(x, W, xw);
  tgat_attn<<<NV, 256, 0, stream>>>(xw, alpha, phi, (float*)d_out);
}